// PSFNet_9637906612606
// MI455X (gfx1250) — compile-verified
//
#include <hip/hip_runtime.h>
#include <math.h>

// Problem dimensions (from reference)
#define CH   256   // N_DIST channels
#define IMH  256   // H
#define IMW  192   // W
#define KT   129   // taps
#define RAD  64    // KT/2
#define SQ   256   // padded square side
#define PADW 32    // (SQ - IMW)/2

#define BAND 144   // 16 outputs + 128 halo, multiple of 4
#define TPAD 176   // padded taps: index gi+16, gi in [-15,143]

typedef __attribute__((ext_vector_type(2))) float v2f;
typedef __attribute__((ext_vector_type(8))) float v8f;

// Branch-free boundary load: clamp address, zero via mask select.
__device__ __forceinline__ float load_masked(const float* __restrict__ p,
                                             long idx, int coord, int lim) {
    int cc = coord < 0 ? 0 : (coord >= lim ? lim - 1 : coord);
    float v = p[idx + (long)(cc - coord)];   // idx computed with raw coord
    return (coord >= 0 && coord < lim) ? v : 0.0f;
}

// ---------------------------------------------------------------------------
// Depthwise 'SAME' conv along H as banded-Toeplitz GEMM on the fp32 WMMA pipe.
//   out[c,h,w] = sum_{h'} g_c[h'-h+64] * in[c,h',w]
// Workgroup = 4 waves = 16(h) x 64(w) output strip. The 144x64 input band and
// zero-padded taps are staged in LDS so the inner loop is two ds_load_b64 +
// one v_wmma_f32_16x16x4_f32, fully branch-free (EXEC stays all-ones).
// ---------------------------------------------------------------------------
__global__ void __launch_bounds__(128)
convH_wmma(const float* __restrict__ in, const float* __restrict__ taps,
           float* __restrict__ out, int H, int W) {
    __shared__ float sgp[TPAD];          // taps padded: sgp[16+i] = g[i]
    __shared__ float ls_in[BAND * 64];   // input band [row 0..143][col 0..63]

    const int c   = blockIdx.z;
    const int tid = threadIdx.x;
    const int h0  = blockIdx.y * 16;
    const int w00 = blockIdx.x * 64;
    const size_t base = (size_t)c * H * W;

    // Stage padded taps (branch-free)
    for (int i = tid; i < TPAD; i += 128) {
        int gi = i - 16;
        int gc = gi < 0 ? 0 : (gi >= KT ? KT - 1 : gi);
        float v = taps[c * KT + gc];
        sgp[i] = (gi >= 0 && gi < KT) ? v : 0.0f;
    }
    // Stage input band rows h0-64 .. h0+79, cols w00 .. w00+63 (coalesced)
    for (int i = tid; i < BAND * 64; i += 128) {
        const int row = i >> 6;          // 0..143
        const int col = i & 63;
        const int hp  = h0 - RAD + row;
        ls_in[i] = load_masked(in, base + (long)hp * W + (w00 + col), hp, H);
    }
    __syncthreads();

    const int lane = tid & 31;
    const int wave = tid >> 5;           // 0..3 -> 16-col sub-tile
    const int half = lane >> 4;
    const int mn   = lane & 15;
    const int w0l  = wave * 16;

    v8f acc = {};
#pragma unroll 4
    for (int kk = 0; kk < 36; ++kk) {
        v2f a, b;
#pragma unroll
        for (int t = 0; t < 2; ++t) {
            const int row = kk * 4 + half * 2 + t;     // band row 0..143
            a[t] = sgp[16 + row - mn];                 // Toeplitz A[m][k]
            b[t] = ls_in[(row << 6) + w0l + mn];       // input   B[k][n]
        }
        acc = __builtin_amdgcn_wmma_f32_16x16x4_f32(
                  false, a, false, b, (short)0, acc, false, false);
    }
#pragma unroll
    for (int r = 0; r < 8; ++r)   // D: VGPR r -> rows r (lanes<16) / r+8
        out[base + (size_t)(h0 + half * 8 + r) * W + (w00 + w0l + mn)] = acc[r];
}

// ---------------------------------------------------------------------------
// Depthwise 'SAME' conv along W: out[c,h,w] = sum_{w'} in[c,h,w'] g_c[w'-w+64]
// Workgroup = 4 waves = 64(h) x 16(w) output strip; 64x144 input band in LDS.
// A = input tile (M=h, K=w' band), B = Toeplitz^T from padded LDS taps.
// ---------------------------------------------------------------------------
__global__ void __launch_bounds__(128)
convW_wmma(const float* __restrict__ in, const float* __restrict__ taps,
           float* __restrict__ out, int H, int W) {
    __shared__ float sgp[TPAD];
    __shared__ float ls_in[64 * BAND];   // [row r 0..63][band col j 0..143]

    const int c   = blockIdx.z;
    const int tid = threadIdx.x;
    const int h00 = blockIdx.y * 64;
    const int w0  = blockIdx.x * 16;
    const size_t base = (size_t)c * H * W;

    for (int i = tid; i < TPAD; i += 128) {
        int gi = i - 16;
        int gc = gi < 0 ? 0 : (gi >= KT ? KT - 1 : gi);
        float v = taps[c * KT + gc];
        sgp[i] = (gi >= 0 && gi < KT) ? v : 0.0f;
    }
    for (int i = tid; i < 64 * BAND; i += 128) {
        const int r  = i / BAND;         // 0..63
        const int j  = i - r * BAND;     // 0..143
        const int wp = w0 - RAD + j;
        ls_in[i] = load_masked(in, base + (long)(h00 + r) * W + wp, wp, W);
    }
    __syncthreads();

    const int lane = tid & 31;
    const int wave = tid >> 5;           // 0..3 -> 16-row sub-tile
    const int half = lane >> 4;
    const int mn   = lane & 15;
    const int r0   = wave * 16;

    v8f acc = {};
#pragma unroll 4
    for (int kk = 0; kk < 36; ++kk) {
        v2f a, b;
#pragma unroll
        for (int t = 0; t < 2; ++t) {
            const int j = kk * 4 + half * 2 + t;       // band col 0..143
            a[t] = ls_in[(r0 + mn) * BAND + j];        // input   A[m][k]
            b[t] = sgp[16 + j - mn];                   // Toeplitz B[k][n]
        }
        acc = __builtin_amdgcn_wmma_f32_16x16x4_f32(
                  false, a, false, b, (short)0, acc, false, false);
    }
#pragma unroll
    for (int r = 0; r < 8; ++r)
        out[base + (size_t)(h00 + r0 + half * 8 + r) * W + (w0 + mn)] = acc[r];
}

// ---------------------------------------------------------------------------
// Forward rotation: dst (C,SQ,SQ) = bilinear-rotate of virtually padded src
// (C,IMH,IMW placed at columns [PADW, PADW+IMW)), zero outside.
// ---------------------------------------------------------------------------
__global__ void rotate_fwd(const float* __restrict__ src,
                           float* __restrict__ dst, float ca, float sa) {
    const size_t total = (size_t)CH * SQ * SQ;
    const size_t tid = (size_t)blockIdx.x * blockDim.x + threadIdx.x;
    if (tid >= total) return;
    const int x = (int)(tid % SQ);
    const int y = (int)((tid / SQ) % SQ);
    const int c = (int)(tid / ((size_t)SQ * SQ));

    const float ctr = (SQ - 1) * 0.5f;
    const float yy = y - ctr, xx = x - ctr;
    const float sy = ca * yy + sa * xx + ctr;
    const float sx = -sa * yy + ca * xx + ctr;
    const float y0f = floorf(sy), x0f = floorf(sx);
    const int y0 = (int)y0f, x0 = (int)x0f;
    const float fy = sy - y0f, fx = sx - x0f;

    auto samp = [&](int yi, int xi) -> float {
        const int xs = xi - PADW;
        const bool ok = (yi >= 0) & (yi < SQ) & (xs >= 0) & (xs < IMW);
        const int yc = yi < 0 ? 0 : (yi >= IMH ? IMH - 1 : yi);
        const int xc = xs < 0 ? 0 : (xs >= IMW ? IMW - 1 : xs);
        const float v = src[((size_t)c * IMH + yc) * IMW + xc];
        return ok ? v : 0.0f;
    };
    dst[tid] = samp(y0,     x0    ) * (1.0f - fy) * (1.0f - fx)
             + samp(y0,     x0 + 1) * (1.0f - fy) * fx
             + samp(y0 + 1, x0    ) * fy          * (1.0f - fx)
             + samp(y0 + 1, x0 + 1) * fy          * fx;
}

// ---------------------------------------------------------------------------
// Inverse rotation + crop + accumulate into acc (C,IMH,IMW).
// Caller passes (cos(-ang), sin(-ang)); src is the full (C,SQ,SQ) square.
// ---------------------------------------------------------------------------
__global__ void rotate_bwd_acc(const float* __restrict__ src,
                               float* __restrict__ acc, float ca, float sa) {
    const size_t total = (size_t)CH * IMH * IMW;
    const size_t tid = (size_t)blockIdx.x * blockDim.x + threadIdx.x;
    if (tid >= total) return;
    const int w = (int)(tid % IMW);
    const int h = (int)((tid / IMW) % IMH);
    const int c = (int)(tid / ((size_t)IMH * IMW));
    const int x = w + PADW;              // position inside the padded square

    const float ctr = (SQ - 1) * 0.5f;
    const float yy = h - ctr, xx = x - ctr;
    const float sy = ca * yy + sa * xx + ctr;
    const float sx = -sa * yy + ca * xx + ctr;
    const float y0f = floorf(sy), x0f = floorf(sx);
    const int y0 = (int)y0f, x0 = (int)x0f;
    const float fy = sy - y0f, fx = sx - x0f;

    auto samp = [&](int yi, int xi) -> float {
        const bool ok = (yi >= 0) & (yi < SQ) & (xi >= 0) & (xi < SQ);
        const int yc = yi < 0 ? 0 : (yi >= SQ ? SQ - 1 : yi);
        const int xc = xi < 0 ? 0 : (xi >= SQ ? SQ - 1 : xi);
        const float v = src[((size_t)c * SQ + yc) * SQ + xc];
        return ok ? v : 0.0f;
    };
    acc[tid] += samp(y0,     x0    ) * (1.0f - fy) * (1.0f - fx)
              + samp(y0,     x0 + 1) * (1.0f - fy) * fx
              + samp(y0 + 1, x0    ) * fy          * (1.0f - fx)
              + samp(y0 + 1, x0 + 1) * fy          * fx;
}

// out = (blurred + (iso + tails accumulated in out)) / norm[c]
__global__ void finalize_k(const float* __restrict__ blurred,
                           const float* __restrict__ norm,
                           float* __restrict__ out) {
    const size_t total = (size_t)CH * IMH * IMW;
    const size_t tid = (size_t)blockIdx.x * blockDim.x + threadIdx.x;
    if (tid >= total) return;
    const int c = (int)(tid / ((size_t)IMH * IMW));
    out[tid] = (blurred[tid] + out[tid]) / norm[c];
}

// ---------------------------------------------------------------------------
extern "C" void kernel_launch(void* const* d_in, const int* in_sizes, int n_in,
                              void* d_out, int out_size, void* d_ws, size_t ws_size,
                              hipStream_t stream) {
    const float* x    = (const float*)d_in[0];   // (1,256,256,192)
    const float* kg   = (const float*)d_in[1];   // (256,129)
    const float* kw   = (const float*)d_in[2];   // (256,129)
    const float* kiso = (const float*)d_in[3];   // (256,129)
    const float* norm = (const float*)d_in[4];   // (256,)
    float* out = (float*)d_out;                  // (1,256,256,192) -> acc+result

    const size_t nimg = (size_t)CH * IMH * IMW;  // 12.58M floats
    const size_t nsq  = (size_t)CH * SQ * SQ;    // 16.78M floats
    float* ws   = (float*)d_ws;                  // needs 2*nimg + 2*nsq floats
    float* tmp  = ws;                            // intermediate image
    float* bg   = ws + nimg;                     // double-gaussian-blurred
    float* rot  = ws + 2 * nimg;                 // rotated square
    float* rotc = rot + nsq;                     // conv'd rotated square

    const dim3 blk(128);                         // 4 wave32 per workgroup
    const dim3 gHI(IMW / 64, IMH / 16, CH);      // convH on image
    const dim3 gWI(IMW / 16, IMH / 64, CH);      // convW on image
    const dim3 gWS(SQ / 16,  SQ / 64,  CH);      // convW on square
    const int  et = 256;
    const unsigned gSqE  = (unsigned)((nsq  + et - 1) / et);
    const unsigned gImgE = (unsigned)((nimg + et - 1) / et);

    // blurred = dwconv_W(dwconv_H(x, g), g) via separable Toeplitz-GEMMs
    convH_wmma<<<gHI, blk, 0, stream>>>(x,   kg,   tmp, IMH, IMW);
    convW_wmma<<<gWI, blk, 0, stream>>>(tmp, kg,   bg,  IMH, IMW);
    // iso path; write straight into d_out (initializes the accumulator)
    convH_wmma<<<gHI, blk, 0, stream>>>(bg,  kiso, tmp, IMH, IMW);
    convW_wmma<<<gWI, blk, 0, stream>>>(tmp, kiso, out, IMH, IMW);

    // tails: 3 x (rotate -> conv_W(kernel_w) -> unrotate-accumulate)
    for (int i = 0; i < 3; ++i) {
        const double ang = (double)i * (M_PI / 3.0);   // 0, 60, 120 deg
        const float ca = (float)cos(ang);
        const float sa = (float)sin(ang);
        rotate_fwd<<<gSqE, et, 0, stream>>>(bg, rot, ca, sa);
        convW_wmma<<<gWS, blk, 0, stream>>>(rot, kw, rotc, SQ, SQ);
        rotate_bwd_acc<<<gImgE, et, 0, stream>>>(rotc, out, ca, -sa);
    }

    finalize_k<<<gImgE, et, 0, stream>>>(bg, norm, out);
}